// FastSpeech2Pros_14989435863124
// MI455X (gfx1250) — compile-verified
//
#include <hip/hip_runtime.h>

// ---------------------------------------------------------------------------
// FastSpeech2 duration-regulator segment-mean as a 0/1-membership matmul:
//   sums[b] (S x C) = Member[b] (S x T, f32 0/1) @ frame_sum[b] (T x C)
// via V_WMMA_F32_16X16X4_F32 (exact f32 accumulate).
//
// MI455X reasoning: pure streaming problem (64 MB read @ 23.3 TB/s => ~2.8us
// floor). Little's law needs ~16 MB in flight, so: b128-only non-temporal
// global loads (channel-permuted WMMA columns so each lane owns 4 contiguous
// channels), 2048 waves (8 waves/block: 4-way time interleave x 2-way frame
// split), LDS combine of partial tiles, b128 coalesced stores. The entire
// time loop is scalarized via readfirstlane (bounds AND start), so EXEC is
// all-ones by construction at every WMMA (ISA requirement) and the loop
// control is pure SALU.
// ---------------------------------------------------------------------------

typedef __attribute__((ext_vector_type(2))) float v2f;
typedef __attribute__((ext_vector_type(4))) float v4f;
typedef __attribute__((ext_vector_type(8))) float v8f;

#define BH  8      // batches used (B/2)
#define T_  1024
#define F_  8
#define C_  256
#define S_  128

// Per-batch inclusive prefix sum of durations (int64 in, int32 out).
__global__ void fs2_scan_kernel(const long long* __restrict__ d,
                                int* __restrict__ cum) {
  __shared__ int buf[S_];
  const int b = blockIdx.x;
  const int s = threadIdx.x;
  buf[s] = (int)d[(size_t)b * S_ + s];
  __syncthreads();
#pragma unroll
  for (int off = 1; off < S_; off <<= 1) {
    int v = buf[s];
    int add = (s >= off) ? buf[s - off] : 0;
    __syncthreads();
    buf[s] = v + add;
    __syncthreads();
  }
  cum[(size_t)b * S_ + s] = buf[s];
}

// grid = (chanTiles=4 [64 ch each], segTiles=8, batches=8), block = 256 (8 waves).
// Wave w: time-interleave slot (w&3), frame half (w>>2). Each wave keeps four
// 16x16 f32 WMMA accumulators; sub-tile j covers channels c0 + 4*N + j.
__global__ __launch_bounds__(256) void
fs2_segmean_wmma_kernel(const float* __restrict__ e,
                        const int* __restrict__ cum,
                        float* __restrict__ out) {
  const int ct   = blockIdx.x;            // 64-channel tile
  const int st   = blockIdx.y;            // 16-segment tile
  const int b    = blockIdx.z;
  const int tid  = threadIdx.x;
  const int w    = tid >> 5;              // wave 0..7
  const int lane = tid & 31;
  const int half = lane >> 4;
  const int l15  = lane & 15;
  const int tsl  = w & 3;                 // time-interleave slot (wave-uniform)
  const int fb   = (w >> 2) * 4;          // frame base: 0 or 4 (wave-uniform)

  __shared__ int   sStart[16];
  __shared__ int   sEnd[16];
  __shared__ float scaleSh[16];
  __shared__ v4f   part[8][8][32];        // [wave][vgpr r][lane] = 4 ch, 32 KB

  const int* cumB = cum + b * S_;
  if (tid < 16) {
    const int s  = st * 16 + tid;
    const int lo = (s == 0) ? 0 : cumB[s - 1];
    const int hi = cumB[s];
    sStart[tid] = lo;
    sEnd[tid]   = hi;
    const int len = hi - lo;
    scaleSh[tid] = (len > 0) ? (1.0f / (float)(len * F_)) : 0.0f;
  }
  __syncthreads();

  const int myS = sStart[l15];            // A-matrix row (seg) = lane&15
  const int myE = sEnd[l15];
  // Fully scalar loop control: start AND bound via readfirstlane (tsl is
  // wave-uniform), so the backend emits an SALU loop and never touches EXEC.
  const int tbStart = __builtin_amdgcn_readfirstlane(sStart[0] + 4 * tsl);
  int t1 = __builtin_amdgcn_readfirstlane(sEnd[15]);
  if (t1 > T_) t1 = T_;

  const int c0 = ct * 64;
  // This lane's 4 contiguous channels: c0 + 4*l15 .. +3 (16B aligned).
  const float* eB = e + (size_t)b * T_ * F_ * C_ + c0 + 4 * l15;

  v8f acc0 = {}, acc1 = {}, acc2 = {}, acc3 = {};

  for (int tb = tbStart; tb < t1; tb += 16) {
    // A (16x4 f32): lane holds M=lane&15, K = 2*half + {0,1}.
    const int tA = tb + 2 * half;
    const int tB = tA + 1;
    const float a0 = (tA >= myS && tA < myE && tA < T_) ? 1.0f : 0.0f;
    const float a1 = (tB >= myS && tB < myE && tB < T_) ? 1.0f : 0.0f;
    v2f A; A[0] = a0; A[1] = a1;

    // Frame sums (this wave's 4 frames) for 4 channels, two time steps.
    // Clamped addresses for t>=T are harmless: membership column is zero.
    // Non-temporal: e is streamed exactly once, keep it out of L2 residency.
    const int tAc = (tA < T_ - 1) ? tA : (T_ - 1);
    const int tBc = (tB < T_ - 1) ? tB : (T_ - 1);
    const v4f* pA = (const v4f*)(eB + ((size_t)tAc * F_ + fb) * C_);
    const v4f* pB = (const v4f*)(eB + ((size_t)tBc * F_ + fb) * C_);
    // f-row stride = C_ floats = 64 v4f; 8x global_load_b128 (TH=NT) total.
    const v4f fa = __builtin_nontemporal_load(pA + 0) +
                   __builtin_nontemporal_load(pA + 64) +
                   __builtin_nontemporal_load(pA + 128) +
                   __builtin_nontemporal_load(pA + 192);
    const v4f fbv = __builtin_nontemporal_load(pB + 0) +
                    __builtin_nontemporal_load(pB + 64) +
                    __builtin_nontemporal_load(pB + 128) +
                    __builtin_nontemporal_load(pB + 192);

    v2f B0; B0[0] = fa[0]; B0[1] = fbv[0];
    v2f B1; B1[0] = fa[1]; B1[1] = fbv[1];
    v2f B2; B2[0] = fa[2]; B2[1] = fbv[2];
    v2f B3; B3[0] = fa[3]; B3[1] = fbv[3];

    acc0 = __builtin_amdgcn_wmma_f32_16x16x4_f32(false, A, false, B0,
                                                 (short)0, acc0, false, false);
    acc1 = __builtin_amdgcn_wmma_f32_16x16x4_f32(false, A, false, B1,
                                                 (short)0, acc1, false, false);
    acc2 = __builtin_amdgcn_wmma_f32_16x16x4_f32(false, A, false, B2,
                                                 (short)0, acc2, false, false);
    acc3 = __builtin_amdgcn_wmma_f32_16x16x4_f32(false, A, false, B3,
                                                 (short)0, acc3, false, false);
  }

  // Stash this wave's partial tile in LDS (b128 DS stores).
#pragma unroll
  for (int r = 0; r < 8; ++r) {
    v4f pv;
    pv[0] = acc0[r]; pv[1] = acc1[r]; pv[2] = acc2[r]; pv[3] = acc3[r];
    part[w][r][lane] = pv;
  }
  __syncthreads();

  // Combine 8 partials, scale by 1/(d*F), store b128 (4 consecutive channels).
  const int rr = (tid >> 5) & 7;          // C/D VGPR index of producers
  const int ln = tid & 31;                // producer lane
  v4f sum = {};
#pragma unroll
  for (int ww = 0; ww < 8; ++ww) sum += part[ww][rr][ln];

  const int seg = rr + 8 * (ln >> 4);     // C/D layout: M = r + 8*(lane>=16)
  const float sc = scaleSh[seg];
  sum *= sc;

  float* outp = out + ((size_t)b * S_ + st * 16 + seg) * C_ + c0 + 4 * (ln & 15);
  *(v4f*)outp = sum;
}

extern "C" void kernel_launch(void* const* d_in, const int* in_sizes, int n_in,
                              void* d_out, int out_size, void* d_ws, size_t ws_size,
                              hipStream_t stream) {
  (void)in_sizes; (void)n_in; (void)out_size; (void)ws_size;
  const float*     e   = (const float*)d_in[0];      // (16,1024,8,256) f32
  const long long* dur = (const long long*)d_in[1];  // (16,128) i64
  float*           out = (float*)d_out;              // (8,128,256) f32
  int*             cum = (int*)d_ws;                 // (8,128) i32 scratch

  fs2_scan_kernel<<<dim3(BH), dim3(S_), 0, stream>>>(dur, cum);
  fs2_segmean_wmma_kernel<<<dim3(C_ / 64, S_ / 16, BH), dim3(256), 0, stream>>>(
      e, cum, out);
}